// BoundaryLoss_62302795596057
// MI455X (gfx1250) — compile-verified
//
#include <hip/hip_runtime.h>
#include <hip/hip_bf16.h>

typedef __attribute__((ext_vector_type(2))) float v2f;
typedef __attribute__((ext_vector_type(8))) float v8f;

namespace {
constexpr int   Bc = 8, Kc = 4, Hc = 256, Wc = 256;
constexpr int   NTOT  = Bc * Kc * Hc * Wc;   // 2,097,152
constexpr int   NPART = Bc * Kc * Hc;        // 8,192 row partials
constexpr float INFV  = 1e10f;               // same sentinel as reference
}

// ---------------------------------------------------------------------------
// Kernel 0: zero the 32 per-(b,k) "class present" flags (ws is poisoned).
// ---------------------------------------------------------------------------
__global__ __launch_bounds__(32) void bl_zero_flags(int* flags) {
  flags[threadIdx.x] = 0;
}

// ---------------------------------------------------------------------------
// Kernel 1: 1-D squared EDT along H (columns), for pos and neg masks at once.
// One block per (b,k,w) column; thread = h. f is binary {0, INFV} exactly as
// in the reference, so g1[h] = min_j f[j] + (h-j)^2 in f32.
// ---------------------------------------------------------------------------
__global__ __launch_bounds__(256) void bl_pass1(const int* __restrict__ y,
                                                float* __restrict__ g1p,
                                                float* __restrict__ g1n,
                                                int* __restrict__ flags) {
  const int blk = blockIdx.x;          // (b*Kc + k)*Wc + w
  const int w   = blk % Wc;
  const int bk  = blk / Wc;
  const int k   = bk % Kc;
  const int b   = bk / Kc;
  const int h   = threadIdx.x;

  __shared__ float fpos[Hc];
  __shared__ float fneg[Hc];
  __shared__ int   anyPos;

  if (h == 0) anyPos = 0;
  __syncthreads();

  const int  yv  = y[(b * Hc + h) * Wc + w];
  const bool pos = (yv == k);
  fpos[h] = pos ? 0.0f : INFV;
  fneg[h] = pos ? INFV : 0.0f;
  if (pos) anyPos = 1;                 // benign race: all writers store 1
  __syncthreads();

  float gp = 3.4e38f, gn = 3.4e38f;
#pragma unroll 8
  for (int j = 0; j < Hc; ++j) {
    const float d  = (float)(h - j);
    const float d2 = d * d;
    gp = fminf(gp, fpos[j] + d2);
    gn = fminf(gn, fneg[j] + d2);
  }

  const size_t o = ((size_t)bk * Hc + h) * Wc + w;
  g1p[o] = gp;
  g1n[o] = gn;

  if (h == 0 && anyPos) atomicOr(&flags[bk], 1);
}

// ---------------------------------------------------------------------------
// Kernel 2: 1-D squared EDT along W (rows) on the pass-1 results, then fuse
// the signed distance map, the x* product, and a per-row reduction.
// One block per (b,k,h) row; thread = w. Fully coalesced global accesses.
// ---------------------------------------------------------------------------
__global__ __launch_bounds__(256) void bl_pass2(const float* __restrict__ x,
                                                const int* __restrict__ y,
                                                const float* __restrict__ g1p,
                                                const float* __restrict__ g1n,
                                                const int* __restrict__ flags,
                                                float* __restrict__ partials) {
  const int blk = blockIdx.x;          // (b*Kc + k)*Hc + h
  const int h   = blk % Hc;
  const int bk  = blk / Hc;
  const int k   = bk % Kc;
  const int b   = bk / Kc;
  const int w   = threadIdx.x;

  __shared__ float sp[Wc];
  __shared__ float sn[Wc];
  __shared__ float red[Wc];

  const size_t rowo = ((size_t)bk * Hc + h) * Wc;
  sp[w] = g1p[rowo + w];
  sn[w] = g1n[rowo + w];
  __syncthreads();

  float gp = 3.4e38f, gn = 3.4e38f;
#pragma unroll 8
  for (int j = 0; j < Wc; ++j) {
    const float d  = (float)(w - j);
    const float d2 = d * d;
    gp = fminf(gp, sp[j] + d2);
    gn = fminf(gn, sn[j] + d2);
  }

  const int  yv  = y[(b * Hc + h) * Wc + w];
  const bool pos = (yv == k);
  const float d_to_pos = sqrtf(gp);    // _edt(pos mask)
  const float d_to_neg = sqrtf(gn);    // _edt(neg mask)
  // res = where(neg, d_to_pos, 0) - where(pos, d_to_neg - 1, 0)
  float res = pos ? (1.0f - d_to_neg) : d_to_pos;
  if (flags[bk] == 0) res = 0.0f;      // class absent -> zero map

  red[w] = x[rowo + w] * res;
  __syncthreads();
#pragma unroll
  for (int off = 128; off > 0; off >>= 1) {
    if (w < off) red[w] += red[w + off];
    __syncthreads();
  }
  if (w == 0) partials[blk] = red[0];
}

// ---------------------------------------------------------------------------
// Kernel 3: deterministic WMMA reduction of the 8192 row partials.
// B = ones, so D[m][n] = sum_k A[m][k] + C[m][n]: each v_wmma_f32_16x16x4_f32
// folds 64 f32 values into the accumulator. Single wave32, EXEC all ones.
// Total = (sum of acc rows at lane 0, M=0..7) + (lane 16, M=8..15), column N=0.
// ---------------------------------------------------------------------------
__global__ __launch_bounds__(32) void bl_finalize(const float* __restrict__ partials,
                                                  float* __restrict__ out) {
  const int lane = threadIdx.x;        // 0..31, wave32
  v8f acc = {};
  v2f ones;
  ones.x = 1.0f;
  ones.y = 1.0f;

  for (int base = 0; base < NPART; base += 64) {
    v2f a;
    a.x = partials[base + lane];
    a.y = partials[base + 32 + lane];
    acc = __builtin_amdgcn_wmma_f32_16x16x4_f32(
        /*neg_a=*/false, a, /*neg_b=*/false, ones,
        /*c_mod=*/(short)0, acc, /*reuse_a=*/false, /*reuse_b=*/false);
  }

  float s = acc[0] + acc[1] + acc[2] + acc[3] +
            acc[4] + acc[5] + acc[6] + acc[7];
  const float s0  = __shfl(s, 0, 32);   // rows M=0..7,  N=0
  const float s16 = __shfl(s, 16, 32);  // rows M=8..15, N=0
  if (lane == 0) out[0] = (s0 + s16) * (1.0f / (float)NTOT);
}

// ---------------------------------------------------------------------------
extern "C" void kernel_launch(void* const* d_in, const int* in_sizes, int n_in,
                              void* d_out, int out_size, void* d_ws, size_t ws_size,
                              hipStream_t stream) {
  const float* x = (const float*)d_in[0];
  const int*   y = (const int*)d_in[1];

  float* ws       = (float*)d_ws;
  float* g1p      = ws;                       // 2,097,152 f32 (8 MB)
  float* g1n      = ws + NTOT;                // 2,097,152 f32 (8 MB)
  float* partials = ws + 2 * (size_t)NTOT;    // 8,192 f32
  int*   flags    = (int*)(ws + 2 * (size_t)NTOT + NPART);  // 32 i32

  float* out = (float*)d_out;

  bl_zero_flags<<<1, 32, 0, stream>>>(flags);
  bl_pass1<<<Bc * Kc * Wc, Hc, 0, stream>>>(y, g1p, g1n, flags);
  bl_pass2<<<Bc * Kc * Hc, Wc, 0, stream>>>(x, y, g1p, g1n, flags, partials);
  bl_finalize<<<1, 32, 0, stream>>>(partials, out);
}